// SoftDTW_62783831933675
// MI455X (gfx1250) — compile-verified
//
#include <hip/hip_runtime.h>

typedef __attribute__((ext_vector_type(2))) float v2f;
typedef __attribute__((ext_vector_type(8))) float v8f;

#define NROWS 2048
#define MCOLS 2048
#define KDIM  64
#define INF_F 1e8f

// ---------------------------------------------------------------------------
// Kernel 1: per-row squared norms of x and y (2048 rows x 64 f32 each)
// ---------------------------------------------------------------------------
__global__ void softdtw_norms(const float* __restrict__ x,
                              const float* __restrict__ y,
                              float* __restrict__ x2,
                              float* __restrict__ y2) {
    int i = blockIdx.x * blockDim.x + threadIdx.x;
    if (i >= NROWS) return;
    const float4* xp = (const float4*)(x + i * KDIM);
    const float4* yp = (const float4*)(y + i * KDIM);
    float sx = 0.f, sy = 0.f;
#pragma unroll
    for (int k = 0; k < KDIM / 4; ++k) {
        float4 a = xp[k];
        float4 b = yp[k];
        sx += a.x * a.x + a.y * a.y + a.z * a.z + a.w * a.w;
        sy += b.x * b.x + b.y * b.y + b.z * b.z + b.w * b.w;
    }
    x2[i] = sx;
    y2[i] = sy;
}

// ---------------------------------------------------------------------------
// Kernel 2: D[i][j] = x2[i] + y2[j] - 2 * (x @ y^T)[i][j]  via f32 WMMA.
// One wave32 computes one 16x16 tile of D, K-loop of 16x16x4 f32 WMMAs.
// ---------------------------------------------------------------------------
__global__ void __launch_bounds__(256) softdtw_dist(const float* __restrict__ x,
                                                    const float* __restrict__ y,
                                                    const float* __restrict__ x2,
                                                    const float* __restrict__ y2,
                                                    float* __restrict__ D) {
    const int lane   = threadIdx.x & 31;
    const int waveId = blockIdx.x * (blockDim.x >> 5) + (threadIdx.x >> 5);
    const int tileM  = (waveId >> 7) << 4;     // waveId / 128 tiles of 16 rows
    const int tileN  = (waveId & 127) << 4;    // waveId % 128 tiles of 16 cols
    const int half   = lane >> 4;              // 0: lanes 0-15, 1: lanes 16-31
    const int lid    = lane & 15;

    // ISA layout: A lane holds M=lid, K = k0 + half*2 (+1 in second VGPR);
    //             B lane holds N=lid, K = k0 + half*2 (+1 in second VGPR).
    const float* xp = x + (tileM + lid) * KDIM + half * 2;
    const float* yp = y + (tileN + lid) * KDIM + half * 2;

    v8f acc = {};
#pragma unroll
    for (int k0 = 0; k0 < KDIM; k0 += 4) {
        v2f a = *(const v2f*)(xp + k0);
        v2f b = *(const v2f*)(yp + k0);
        acc = __builtin_amdgcn_wmma_f32_16x16x4_f32(
            /*neg_a=*/false, a, /*neg_b=*/false, b,
            /*c_mod=*/(short)0, acc, /*reuse_a=*/false, /*reuse_b=*/false);
    }

    // C/D layout: VGPR r, lane l -> M = r + half*8, N = lid
    const int cN  = tileN + lid;
    const float yn2 = y2[cN];
#pragma unroll
    for (int r = 0; r < 8; ++r) {
        const int cM = tileM + r + half * 8;
        D[cM * MCOLS + cN] = x2[cM] + yn2 - 2.0f * acc[r];
    }
}

// ---------------------------------------------------------------------------
// Kernel 3: soft-DTW anti-diagonal DP (gamma = 1).
// Single 1024-thread workgroup (32 wave32s); three rolling diagonals in LDS.
// R[i,j] = D[i-1,j-1] + softmin(R[i-1,j-1], R[i-1,j], R[i,j-1])
// ---------------------------------------------------------------------------
__global__ void __launch_bounds__(1024) softdtw_dp(const float* __restrict__ D,
                                                   float* __restrict__ out) {
    constexpr int L = NROWS + 1;               // 2049 entries per diagonal
    __shared__ float buf[3][L];
    const int t = threadIdx.x;

    for (int i = t; i < L; i += 1024) {
        buf[0][i] = (i == 0) ? 0.0f : INF_F;   // diagonal d=0
        buf[1][i] = INF_F;                     // diagonal d=1
    }
    __syncthreads();

    for (int d = 2; d <= NROWS + MCOLS; ++d) {
        const float* prev2 = buf[(d - 2) % 3];
        const float* prev1 = buf[(d - 1) % 3];
        float*       cur   = buf[d % 3];
        for (int i = t; i < L; i += 1024) {
            const int j = d - i;
            float v = INF_F;
            if (i >= 1 && i <= NROWS && j >= 1 && j <= MCOLS) {
                const float* dp = D + (i - 1) * MCOLS + (j - 1);
                __builtin_prefetch(dp + 32, 0, 1);   // next lines of this row
                const float dv = *dp;
                const float a = prev2[i - 1];        // R[i-1, j-1]
                const float b = prev1[i - 1];        // R[i-1, j]
                const float c = prev1[i];            // R[i,   j-1]
                const float m = fminf(a, fminf(b, c));
                const float s = __expf(m - a) + __expf(m - b) + __expf(m - c);
                v = dv + m - __logf(s);              // gamma == 1
            }
            cur[i] = v;
        }
        __syncthreads();
    }

    if (t == 0) out[0] = buf[(NROWS + MCOLS) % 3][NROWS];
}

// ---------------------------------------------------------------------------
extern "C" void kernel_launch(void* const* d_in, const int* in_sizes, int n_in,
                              void* d_out, int out_size, void* d_ws, size_t ws_size,
                              hipStream_t stream) {
    const float* x = (const float*)d_in[0];   // [2048, 64] f32
    const float* y = (const float*)d_in[1];   // [2048, 64] f32

    float* D  = (float*)d_ws;                 // 2048*2048 f32 = 16 MB
    float* x2 = D + (size_t)NROWS * MCOLS;    // 2048 f32
    float* y2 = x2 + NROWS;                   // 2048 f32

    softdtw_norms<<<(NROWS + 255) / 256, 256, 0, stream>>>(x, y, x2, y2);

    // 128x128 tiles of 16x16, one wave per tile; 8 waves (256 threads) / block
    const int totalWaves = 128 * 128;
    softdtw_dist<<<totalWaves / 8, 256, 0, stream>>>(x, y, x2, y2, D);

    softdtw_dp<<<1, 1024, 0, stream>>>(D, (float*)d_out);
}